// ParticleNet_2542620639809
// MI455X (gfx1250) — compile-verified
//
#include <hip/hip_runtime.h>

typedef float v2f __attribute__((ext_vector_type(2)));
typedef float v8f __attribute__((ext_vector_type(8)));

#define GRP   256
#define PTS   128
#define KNN   16
#define NPTS  (GRP * PTS)      // 32768
#define NEDGE (NPTS * KNN)     // 524288
#define TILE_K 32
#define ASTR  34               // A LDS row stride (conflict-free both lane halves)
#define BPSTR 160              // B pair-row stride in floats (conflict-free b64 reads)

// ------------------------------------------------------------------
// GEMM: Y[N,M] = f(A[N,K]) @ W[K,M] (row-major), f32 WMMA 16x16x4.
// Block tile 128x64, 8 waves x (16 rows x 64 cols), 4 accumulators/wave.
// A and B fragments are stored "fragment-ready" so each WMMA operand is
// one ds_load_b64 into an even-aligned VGPR pair (no repack movs).
// AMODE=1: f(a)=relu((a-mu)*sc+bs) fused on A load (prev layer's BN).
// EPI=1:   y = relu(y + bias).
// STATS=1: per-column sum / sum-of-squares of the stored output are
//          reduced in-block and atomically accumulated (fused BN stats).
// Requires N%128==0, M%64==0; K arbitrary (zero-padded chunks).
// ------------------------------------------------------------------
template <int AMODE, int EPI, int STATS>
__global__ __launch_bounds__(256) void gemm_wmma(
    const float* __restrict__ A, const float* __restrict__ W,
    float* __restrict__ Y, int N, int Kdim, int M,
    const float* __restrict__ mu, const float* __restrict__ sc,
    const float* __restrict__ bs, const float* __restrict__ bias,
    float* __restrict__ accS, float* __restrict__ accQ)
{
  __shared__ float As[128][ASTR];            // 128 x 32 (+pad)
  __shared__ float Wp[TILE_K / 2][BPSTR];    // K-pairs x 64 cols, interleaved
  const int tid   = threadIdx.x;
  const int wave  = tid >> 5;
  const int lane  = tid & 31;
  const int row16 = lane & 15;
  const int half  = lane >> 4;               // 0 or 1
  const int khalf = half << 1;               // 0 or 2 (K split across lane halves)
  const long long nbase = (long long)blockIdx.x * 128;
  const int m0 = blockIdx.y * 64;
  v8f acc[4];
  #pragma unroll
  for (int mt = 0; mt < 4; mt++) acc[mt] = v8f{0.f,0.f,0.f,0.f,0.f,0.f,0.f,0.f};

  for (int k0 = 0; k0 < Kdim; k0 += TILE_K) {
    const bool more = (k0 + TILE_K) < Kdim;
    // 128 x TILE_K A-tile, coalesced, BN+ReLU fused.
    for (int e = tid; e < 128 * TILE_K; e += 256) {
      int r = e >> 5;
      int c = e & (TILE_K - 1);
      int kk = k0 + c;
      float a = 0.f;
      if (kk < Kdim) {
        const float* ap = &A[(nbase + r) * (long long)Kdim + kk];
        a = *ap;
        if (more) __builtin_prefetch(ap + TILE_K, 0, 1);   // next K chunk
        if (AMODE == 1) {
          a = (a - mu[kk]) * sc[kk] + bs[kk];
          a = a > 0.f ? a : 0.f;
        }
      }
      As[r][c] = a;
    }
    // TILE_K x 64 W-tile, stored as (k even, k odd) pairs per column.
    for (int e = tid; e < TILE_K * 64; e += 256) {
      int kk = e >> 6;
      int c  = e & 63;
      float w = 0.f;
      if (k0 + kk < Kdim) {
        const float* wp = &W[(long long)(k0 + kk) * M + (m0 + c)];
        w = *wp;
        if (more) __builtin_prefetch(wp + (long long)TILE_K * M, 0, 1);
      }
      Wp[kk >> 1][(c << 1) | (kk & 1)] = w;
    }
    __syncthreads();
    const int r0 = wave * 16;
    #pragma unroll
    for (int kk = 0; kk < TILE_K; kk += 4) {
      // A 16x4 f32: lane%16 = row, v0:{K0|K2}, v1:{K1|K3} -> one b64.
      v2f av = *(const v2f*)&As[r0 + row16][kk + khalf];
      #pragma unroll
      for (int mt = 0; mt < 4; mt++) {
        // B 4x16 f32: lane%16 = col, pair row = (kk+khalf)/2 -> one b64.
        v2f bv = *(const v2f*)&Wp[(kk >> 1) + half][(mt * 16 + row16) << 1];
        acc[mt] = __builtin_amdgcn_wmma_f32_16x16x4_f32(false, av, false, bv,
                                                        (short)0, acc[mt],
                                                        false, false);
      }
    }
    __syncthreads();
  }
  // D layout: VGPR i -> row i (lanes 0-15) / row i+8 (lanes 16-31), col = lane%16
  const int rowadd = half << 3;
  float pS[4], pQ[4];
  #pragma unroll
  for (int mt = 0; mt < 4; mt++) { pS[mt] = 0.f; pQ[mt] = 0.f; }
  #pragma unroll
  for (int mt = 0; mt < 4; mt++) {
    const int colg = m0 + mt * 16 + row16;
    #pragma unroll
    for (int i = 0; i < 8; i++) {
      long long rg = nbase + wave * 16 + rowadd + i;
      float v = acc[mt][i];
      if (EPI == 1) {
        v += bias[colg];
        v = v > 0.f ? v : 0.f;
      }
      Y[rg * (long long)M + colg] = v;
      if (STATS == 1) { pS[mt] += v; pQ[mt] += v * v; }
    }
  }
  if (STATS == 1) {
    // lanes i and i+16 hold the same column -> merge, then cross-wave via LDS.
    float* red = &As[0][0];   // reuse (all waves passed the final barrier)
    #pragma unroll
    for (int mt = 0; mt < 4; mt++) {
      float S = pS[mt] + __shfl_xor(pS[mt], 16, 32);
      float Q = pQ[mt] + __shfl_xor(pQ[mt], 16, 32);
      if (lane < 16) {
        red[wave * 64 + mt * 16 + lane]       = S;
        red[512 + wave * 64 + mt * 16 + lane] = Q;
      }
    }
    __syncthreads();
    if (tid < 64) {
      float S = 0.f, Q = 0.f;
      #pragma unroll
      for (int w = 0; w < 8; w++) {
        S += red[w * 64 + tid];
        Q += red[512 + w * 64 + tid];
      }
      atomicAdd(&accS[m0 + tid], S);
      atomicAdd(&accQ[m0 + tid], Q);
    }
  }
}

// ---------------- kNN (replicates sq_p + sq_q - 2*dot + 1e9*diag) --------
__global__ __launch_bounds__(PTS) void knn_kernel(const float* __restrict__ pts,
                                                  int D, int* __restrict__ idx)
{
  extern __shared__ float PT[];   // transposed [D][128]
  int g = blockIdx.x, p = threadIdx.x;
  const float* base = pts + (long long)g * PTS * D;
  float sq = 0.f;
  for (int d = 0; d < D; d++) {
    float vv = base[(long long)p * D + d];
    PT[d * PTS + p] = vv;
    sq += vv * vv;
  }
  __syncthreads();
  float bd[KNN]; int bi[KNN];
  #pragma unroll
  for (int j = 0; j < KNN; j++) { bd[j] = 3.0e38f; bi[j] = 0; }
  for (int q = 0; q < PTS; q++) {
    float dot = 0.f, sqq = 0.f;
    for (int d = 0; d < D; d++) {
      float a = PT[d * PTS + p];     // conflict-free (stride 1 in p)
      float b = PT[d * PTS + q];     // broadcast
      dot += a * b;
      sqq += b * b;
    }
    float dist = sq + sqq - 2.f * dot;
    if (q == p) dist += 1e9f;
    if (dist < bd[KNN - 1]) {        // stable insertion == top_k tie order
      int pos = KNN - 1;
      while (pos > 0 && bd[pos - 1] > dist) {
        bd[pos] = bd[pos - 1]; bi[pos] = bi[pos - 1]; pos--;
      }
      bd[pos] = dist; bi[pos] = q;
    }
  }
  long long o = ((long long)g * PTS + p) * KNN;
  for (int j = 0; j < KNN; j++) idx[o + j] = bi[j];
}

// -------- y1[node,j,:] = u[node,:] + v[neighbor,:] with fused BN stats ----
// Block covers rowsPer edge-rows; threads: c = tid%cout, row-lane = tid/cout.
__global__ __launch_bounds__(256) void gather_add_stats(
    const float* __restrict__ u, const float* __restrict__ v,
    const int* __restrict__ idx, float* __restrict__ y, int cout,
    long long rowsPer, float* __restrict__ accS, float* __restrict__ accQ)
{
  __shared__ float ls[256], lq[256];
  int tid = threadIdx.x;
  int c = tid % cout;
  int rg = tid / cout;
  int rstep = 256 / cout;
  long long r    = (long long)blockIdx.x * rowsPer + rg;
  long long rend = (long long)(blockIdx.x + 1) * rowsPer;
  if (rend > NEDGE) rend = NEDGE;
  float S = 0.f, Q = 0.f;
  for (; r < rend; r += rstep) {
    long long node = r >> 4;
    long long g    = node >> 7;
    int nb = idx[r];
    float val = u[node * (long long)cout + c] +
                v[(g * PTS + nb) * (long long)cout + c];
    y[r * (long long)cout + c] = val;
    S += val; Q += val * val;
  }
  ls[tid] = S; lq[tid] = Q;
  __syncthreads();
  if (rg == 0) {
    for (int g2 = 1; g2 < rstep; g2++) { S += ls[c + g2 * cout]; Q += lq[c + g2 * cout]; }
    atomicAdd(&accS[c], S);
    atomicAdd(&accQ[c], Q);
  }
}

// ---------------- small-column stats (bn0, M=7) ----------------
__global__ void colstats_small(const float* __restrict__ X, int N, int M,
                               float* __restrict__ accS, float* __restrict__ accQ)
{
  __shared__ float ls[256], lq[256];
  int c = blockIdx.x;
  float s = 0.f, q = 0.f;
  for (int r = threadIdx.x; r < N; r += 256) {
    float v = X[(long long)r * M + c];
    s += v; q += v * v;
  }
  ls[threadIdx.x] = s; lq[threadIdx.x] = q;
  __syncthreads();
  for (int o = 128; o > 0; o >>= 1) {
    if (threadIdx.x < o) { ls[threadIdx.x] += ls[threadIdx.x + o];
                           lq[threadIdx.x] += lq[threadIdx.x + o]; }
    __syncthreads();
  }
  if (threadIdx.x == 0) { accS[c] = ls[0]; accQ[c] = lq[0]; }
}

__global__ void stats_finalize(const float* __restrict__ accS, const float* __restrict__ accQ,
                               const float* __restrict__ g, const float* __restrict__ b,
                               float invN, float* __restrict__ mu, float* __restrict__ sc,
                               float* __restrict__ bsh, int M)
{
  int c = threadIdx.x;
  if (c >= M) return;
  float m   = accS[c] * invN;
  float var = accQ[c] * invN - m * m;
  mu[c]  = m;
  sc[c]  = g[c] * rsqrtf(var + 1e-5f);
  bsh[c] = b[c];
}

__global__ void zero_f32(float* p, int n)
{
  int t = blockIdx.x * 256 + threadIdx.x;
  if (t < n) p[t] = 0.f;
}

// ---------------- misc elementwise kernels ----------------
__global__ void wdiff_kernel(const float* __restrict__ W, float* __restrict__ Wd, int n)
{
  int t = blockIdx.x * 256 + threadIdx.x;
  if (t < n) Wd[t] = W[t] - W[n + t];   // W_top - W_bot
}

__global__ void bn_apply(const float* __restrict__ X, float* __restrict__ Y,
                         const float* __restrict__ mu, const float* __restrict__ s,
                         const float* __restrict__ b, long long tot, int M)
{
  long long t = (long long)blockIdx.x * 256 + threadIdx.x;
  if (t >= tot) return;
  int c = (int)(t % M);
  Y[t] = (X[t] - mu[c]) * s[c] + b[c];
}

// out = relu( max_j relu(bn3(y3)) + bn_s(skip) )
__global__ void combine_kernel(const float* __restrict__ y3, const float* __restrict__ sk,
    const float* __restrict__ mu3, const float* __restrict__ s3, const float* __restrict__ b3,
    const float* __restrict__ mus, const float* __restrict__ ss, const float* __restrict__ bs2,
    float* __restrict__ outp, int cout)
{
  long long t = (long long)blockIdx.x * 256 + threadIdx.x;
  long long tot = (long long)NPTS * cout;
  if (t >= tot) return;
  int c = (int)(t % cout);
  long long node = t / cout;
  float mu_ = mu3[c], s_ = s3[c], b_ = b3[c];
  long long base = node * KNN;
  float m = 0.f;                         // relu'd values are >= 0
  for (int j = 0; j < KNN; j++) {
    float yv = (y3[(base + j) * (long long)cout + c] - mu_) * s_ + b_;
    yv = yv > 0.f ? yv : 0.f;
    m = fmaxf(m, yv);
  }
  float skv = (sk[t] - mus[c]) * ss[c] + bs2[c];
  float r = m + skv;
  outp[t] = r > 0.f ? r : 0.f;
}

__global__ void mean_kernel(const float* __restrict__ fts, float* __restrict__ h, int C)
{
  int g = blockIdx.x, c = threadIdx.x;
  float s = 0.f;
  for (int p = 0; p < PTS; p++) s += fts[((long long)g * PTS + p) * C + c];
  h[g * C + c] = s * (1.f / PTS);
}

__global__ void out_kernel(const float* __restrict__ h, const float* __restrict__ W,
                           const float* __restrict__ b, float* __restrict__ outp)
{
  int g = threadIdx.x;
  float s = 0.f;
  for (int c = 0; c < 256; c++) s += h[g * 256 + c] * W[c];
  s += b[0];
  outp[g] = 1.f / (1.f + expf(-s));
}

// ==================================================================
extern "C" void kernel_launch(void* const* d_in, const int* in_sizes, int n_in,
                              void* d_out, int out_size, void* d_ws, size_t ws_size,
                              hipStream_t stream)
{
  (void)n_in; (void)out_size; (void)ws_size;
  struct ConvW { const float *W1,*g1,*b1,*W2,*g2,*b2,*W3,*g3,*b3,*Ws,*gs,*bs; };
  const float *x, *pos, *bn0g, *bn0b, *fcW, *fcB, *outW, *outB;
  ConvW cw[3];
  auto F = [&](int i) { return (const float*)d_in[i]; };

  if (in_sizes[0] == NPTS * 7) {
    // dict insertion order: x, pos, batch, bn0{g,b}, convs[{mlp{W,g,b}x3, skip{W,g,b}}], fc{W,b}, out{W,b}
    x = F(0); pos = F(1); bn0g = F(3); bn0b = F(4);
    int i = 5;
    for (int c = 0; c < 3; c++) {
      cw[c].W1 = F(i);     cw[c].g1 = F(i + 1);  cw[c].b1 = F(i + 2);
      cw[c].W2 = F(i + 3); cw[c].g2 = F(i + 4);  cw[c].b2 = F(i + 5);
      cw[c].W3 = F(i + 6); cw[c].g3 = F(i + 7);  cw[c].b3 = F(i + 8);
      cw[c].Ws = F(i + 9); cw[c].gs = F(i + 10); cw[c].bs = F(i + 11);
      i += 12;
    }
    fcW = F(i); fcB = F(i + 1); outW = F(i + 2); outB = F(i + 3);
  } else {
    // jax pytree order (dict keys sorted): batch, params{bn0{b,g}, convs[{mlp{W,b,g}x3, skip{W,b,g}}], fc{W,b}, out{W,b}}, pos, x
    bn0b = F(1); bn0g = F(2);
    int i = 3;
    for (int c = 0; c < 3; c++) {
      cw[c].W1 = F(i);     cw[c].b1 = F(i + 1);  cw[c].g1 = F(i + 2);
      cw[c].W2 = F(i + 3); cw[c].b2 = F(i + 4);  cw[c].g2 = F(i + 5);
      cw[c].W3 = F(i + 6); cw[c].b3 = F(i + 7);  cw[c].g3 = F(i + 8);
      cw[c].Ws = F(i + 9); cw[c].bs = F(i + 10); cw[c].gs = F(i + 11);
      i += 12;
    }
    fcW = F(i); fcB = F(i + 1); outW = F(i + 2); outB = F(i + 3);
    pos = F(i + 4); x = F(i + 5);
  }

  // ---- workspace layout ----
  char* base = (char*)d_ws;
  size_t off = 0;
  auto take = [&](size_t bytes) -> float* {
    float* p = (float*)(base + off);
    off = (off + bytes + 511) & ~(size_t)511;
    return p;
  };
  float* yA   = take((size_t)NEDGE * 256 * 4);   // 512 MB
  float* yB   = take((size_t)NEDGE * 256 * 4);   // 512 MB
  float* ftsA = take((size_t)NPTS * 256 * 4);
  float* ftsB = take((size_t)NPTS * 256 * 4);
  float* u    = take((size_t)NPTS * 256 * 4);
  float* v    = take((size_t)NPTS * 256 * 4);
  float* sk   = take((size_t)NPTS * 256 * 4);
  float* wd   = take(256 * 1024);
  int*   idx  = (int*)take((size_t)NEDGE * 4);
  float* st   = take(4096 * 4);
  float* h1   = take(256 * 256 * 4);
  float* h2   = take(256 * 256 * 4);
  float *accS = st, *accQ = st + 256;
  float *muA = st + 512,  *sA = st + 768,  *bA = st + 1024;
  float *muB = st + 1280, *sB = st + 1536, *bB = st + 1792;

  auto zeroacc = [&]() { zero_f32<<<2, 256, 0, stream>>>(accS, 512); };
  auto fin = [&](long long N, int M, const float* g, const float* b,
                 float* mu, float* s, float* bsh) {
    stats_finalize<<<1, 256, 0, stream>>>(accS, accQ, g, b,
                                          (float)(1.0 / (double)N), mu, s, bsh, M);
  };

  // ---- bn0 on x (32768 x 7) ----
  colstats_small<<<7, 256, 0, stream>>>(x, NPTS, 7, accS, accQ);
  fin(NPTS, 7, bn0g, bn0b, muA, sA, bA);
  {
    long long tot = (long long)NPTS * 7;
    bn_apply<<<(int)((tot + 255) / 256), 256, 0, stream>>>(x, ftsA, muA, sA, bA, tot, 7);
  }

  // ---- three EdgeConv blocks ----
  const float* ftsIn = ftsA;
  float* ftsOut = ftsB;
  int cin = 7;
  const int couts[3] = {64, 128, 256};
  for (int c = 0; c < 3; c++) {
    int cout = couts[c];
    const float* ptsP = (c == 0) ? pos : ftsIn;
    int D = (c == 0) ? 2 : cin;
    knn_kernel<<<GRP, PTS, (size_t)PTS * D * 4, stream>>>(ptsP, D, idx);

    { int n = cin * cout;
      wdiff_kernel<<<(n + 255) / 256, 256, 0, stream>>>(cw[c].W1, wd, n); }

    dim3 gU(NPTS / 128, cout / 64);
    gemm_wmma<0, 0, 0><<<gU, 256, 0, stream>>>(ftsIn, wd, u, NPTS, cin, cout,
        nullptr, nullptr, nullptr, nullptr, nullptr, nullptr);
    gemm_wmma<0, 0, 0><<<gU, 256, 0, stream>>>(ftsIn, cw[c].W1 + (size_t)cin * cout, v,
        NPTS, cin, cout, nullptr, nullptr, nullptr, nullptr, nullptr, nullptr);

    // edge layer 1: gather-add with fused stats
    zeroacc();
    { long long rowsPer = 512;
      gather_add_stats<<<(int)(NEDGE / rowsPer), 256, 0, stream>>>(
          u, v, idx, yA, cout, rowsPer, accS, accQ); }
    fin(NEDGE, cout, cw[c].g1, cw[c].b1, muA, sA, bA);

    dim3 gE(NEDGE / 128, cout / 64);
    zeroacc();
    gemm_wmma<1, 0, 1><<<gE, 256, 0, stream>>>(yA, cw[c].W2, yB, NEDGE, cout, cout,
        muA, sA, bA, nullptr, accS, accQ);
    fin(NEDGE, cout, cw[c].g2, cw[c].b2, muA, sA, bA);

    zeroacc();
    gemm_wmma<1, 0, 1><<<gE, 256, 0, stream>>>(yB, cw[c].W3, yA, NEDGE, cout, cout,
        muA, sA, bA, nullptr, accS, accQ);
    fin(NEDGE, cout, cw[c].g3, cw[c].b3, muA, sA, bA);

    zeroacc();
    gemm_wmma<0, 0, 1><<<gU, 256, 0, stream>>>(ftsIn, cw[c].Ws, sk, NPTS, cin, cout,
        nullptr, nullptr, nullptr, nullptr, accS, accQ);
    fin(NPTS, cout, cw[c].gs, cw[c].bs, muB, sB, bB);

    { long long tot = (long long)NPTS * cout;
      combine_kernel<<<(int)((tot + 255) / 256), 256, 0, stream>>>(
          yA, sk, muA, sA, bA, muB, sB, bB, ftsOut, cout); }

    const float* t = ftsIn; ftsIn = ftsOut; ftsOut = (float*)t;
    cin = cout;
  }

  // ---- head: mean pool -> FC+ReLU -> Linear+sigmoid ----
  mean_kernel<<<GRP, 256, 0, stream>>>(ftsIn, h1, 256);
  dim3 gF(256 / 128, 256 / 64);
  gemm_wmma<0, 1, 0><<<gF, 256, 0, stream>>>(h1, fcW, h2, 256, 256, 256,
      nullptr, nullptr, nullptr, fcB, nullptr, nullptr);
  out_kernel<<<1, 256, 0, stream>>>(h2, outW, outB, (float*)d_out);
}